// CapLayer2_26731876450735
// MI455X (gfx1250) — compile-verified
//
#include <hip/hip_runtime.h>
#include <math.h>

// Problem constants (from reference)
#define BS  256   // batch
#define CIN 256   // C, input channels (= GEMM K for delta, M-ish for y)
#define II  256   // I = S*S input capsules
#define JJ  32    // output capsules
#define DD  64    // capsule dim
#define LXS 260   // padded LDS row stride in floats (pad 4 -> bank-conflict-free
                  // float2 fragment reads, 16B-aligned float4 stores)

typedef __attribute__((ext_vector_type(2))) float v2f;
typedef __attribute__((ext_vector_type(8))) float v8f;

// ---------------------------------------------------------------------------
// Kernel 1: softmax over j (axis=1) of routing logits, plus csum[b,j]=sum_i c
// grid (BS), block 256 (thread = i). Fully coalesced (stride over j rows).
// ---------------------------------------------------------------------------
__global__ __launch_bounds__(256)
void cap_softmax(const float* __restrict__ bsrc, float* __restrict__ cbuf,
                 float* __restrict__ csum) {
  __shared__ float part[JJ * 8];
  const int b = blockIdx.x, i = threadIdx.x;
  const float* bp = bsrc + (size_t)b * JJ * II + i;
  float vals[JJ];
  float mx = -3.402823466e38f;
#pragma unroll
  for (int jv = 0; jv < JJ; ++jv) {
    vals[jv] = bp[(size_t)jv * II];
    mx = fmaxf(mx, vals[jv]);
  }
  float sum = 0.f;
#pragma unroll
  for (int jv = 0; jv < JJ; ++jv) {
    vals[jv] = __expf(vals[jv] - mx);
    sum += vals[jv];
  }
  const float inv = 1.f / sum;
  float* cp = cbuf + (size_t)b * JJ * II + i;
  const int wv = i >> 5, ln = i & 31;  // wave32
#pragma unroll
  for (int jv = 0; jv < JJ; ++jv) {
    const float cv = vals[jv] * inv;
    cp[(size_t)jv * II] = cv;
    float r = cv;
    for (int off = 16; off > 0; off >>= 1) r += __shfl_xor(r, off, 32);
    if (ln == 0) part[jv * 8 + wv] = r;
  }
  __syncthreads();
  if (i < JJ) {
    float s = 0.f;
#pragma unroll
    for (int wq = 0; wq < 8; ++wq) s += part[i * 8 + wq];
    csum[b * JJ + i] = s;
  }
}

// ---------------------------------------------------------------------------
// Kernel 2: y[b,j,c] = sum_i c[b,j,i] * x[b,c,i]   (per-batch J x C = JxI @ IxC)
// WMMA f32 16x16x4. Block = (batch b, 64-channel chunk); 8 waves, 1 tile each
// (2 j-tiles x 4 c-tiles). x and c tiles staged in LDS (padded stride).
// ---------------------------------------------------------------------------
__global__ __launch_bounds__(256)
void cap_gemm_y(const float* __restrict__ x, const float* __restrict__ cbuf,
                float* __restrict__ yt) {
  __shared__ float lx[64 * LXS];   // 64 channels x 256 i
  __shared__ float lc[JJ * LXS];   // 32 j x 256 i
  const int b = blockIdx.x;
  const int cc = blockIdx.y * 64;
  const int tid = threadIdx.x;

  const float* xb = x + ((size_t)b * CIN + cc) * II;
#pragma unroll
  for (int r = 0; r < 16; ++r) {   // 16384 floats, coalesced float4
    const int flat = r * 1024 + tid * 4;
    const int row = flat >> 8, col = flat & 255;
    *(float4*)&lx[row * LXS + col] = *(const float4*)(xb + row * II + col);
  }
  const float* cb = cbuf + (size_t)b * JJ * II;
#pragma unroll
  for (int r = 0; r < 8; ++r) {    // 8192 floats
    const int flat = r * 1024 + tid * 4;
    const int row = flat >> 8, col = flat & 255;
    *(float4*)&lc[row * LXS + col] = *(const float4*)(cb + row * II + col);
  }
  __syncthreads();

  const int w = tid >> 5, L = tid & 31, ln = L & 15;
  const int jt = w & 1, ct = w >> 1;
  const int koff = (L >> 4) << 1;  // lanes 16..31 hold K+2,K+3
  const float* arow = &lc[(jt * 16 + ln) * LXS + koff];  // A[m=ln, k..k+1]
  const float* brow = &lx[(ct * 16 + ln) * LXS + koff];  // B[k..k+1, n=ln]
  v8f acc = {};
#pragma unroll 8
  for (int k0 = 0; k0 < II; k0 += 4) {
    const v2f a = *(const v2f*)(arow + k0);
    const v2f bb = *(const v2f*)(brow + k0);
    acc = __builtin_amdgcn_wmma_f32_16x16x4_f32(false, a, false, bb,
                                                (short)0, acc, false, false);
  }
  const int rowj = jt * 16 + ((L >> 4) << 3);  // D vgpr r: M = r (+8 upper half)
  const int cg = cc + ct * 16 + ln;
  float* yb = yt + (size_t)b * JJ * CIN;
#pragma unroll
  for (int r = 0; r < 8; ++r) yb[(rowj + r) * CIN + cg] = acc[r];
}

// ---------------------------------------------------------------------------
// Kernel 3: s[b,j,d] = sum_c W[j,d,c]*y[b,j,c] + bias*csum; v = squash(s);
//           u[b,j,c] = sum_d v[b,j,d]*W[j,d,c]; vb[b,j] = v . bias_j
// grid (J, BS), block 256. Tiny contractions; VALU is fine here.
// ---------------------------------------------------------------------------
__global__ __launch_bounds__(256)
void cap_caps(const float* __restrict__ yt, const float* __restrict__ Wm,
              const float* __restrict__ bias, const float* __restrict__ csum,
              float* __restrict__ vout, float* __restrict__ ubuf,
              float* __restrict__ vb) {
  __shared__ float ly[CIN];
  __shared__ float ps[4 * DD];
  __shared__ float red[DD];
  __shared__ float lv[DD];
  const int j = blockIdx.x, b = blockIdx.y, tid = threadIdx.x;

  ly[tid] = yt[((size_t)b * JJ + j) * CIN + tid];
  __syncthreads();

  const int d = tid & 63, q = tid >> 6;  // 4 partial sums per output d
  const float* wrow = Wm + (size_t)(j * DD + d) * CIN + q * 64;
  const float* lyq = &ly[q * 64];
  float p = 0.f;
#pragma unroll 8
  for (int c = 0; c < 64; ++c) p += wrow[c] * lyq[c];
  ps[q * DD + d] = p;
  __syncthreads();

  float sv = 0.f;
  if (tid < DD) {
    sv = ps[tid] + ps[DD + tid] + ps[2 * DD + tid] + ps[3 * DD + tid] +
         bias[j * DD + tid] * csum[b * JJ + j];
    red[tid] = sv * sv;
  }
  __syncthreads();
  for (int off = 32; off > 0; off >>= 1) {
    if (tid < off) red[tid] += red[tid + off];
    __syncthreads();
  }
  const float n2 = red[0];
  __syncthreads();
  if (tid < DD) {
    const float nrm = sqrtf(n2);
    const float scale = (nrm > 0.f) ? (n2 / (1.f + n2) / nrm) : 0.f;
    const float vd = sv * scale;
    vout[((size_t)b * JJ + j) * DD + tid] = vd;  // final output v[b,j,d]
    lv[tid] = vd;
    red[tid] = vd * bias[j * DD + tid];
  }
  __syncthreads();
  for (int off = 32; off > 0; off >>= 1) {
    if (tid < off) red[tid] += red[tid + off];
    __syncthreads();
  }
  if (tid == 0) vb[b * JJ + j] = red[0];

  // u[b,j,c]: thread = c, W rows read coalesced across threads
  float ua = 0.f;
#pragma unroll 8
  for (int dd = 0; dd < DD; ++dd)
    ua += Wm[(size_t)(j * DD + dd) * CIN + tid] * lv[dd];
  ubuf[((size_t)b * JJ + j) * CIN + tid] = ua;
}

// ---------------------------------------------------------------------------
// Kernel 4: b_dst[b,j,i] = b_src[b,j,i] + sum_c u[b,j,c]*x[b,c,i] + vb[b,j]
// per-batch (J x I) = (J x C) @ (C x I). WMMA f32 16x16x4, u staged in LDS,
// x B-fragments loaded lane-coalesced from global (L2-resident after kernel 2).
// ---------------------------------------------------------------------------
__global__ __launch_bounds__(256)
void cap_delta(const float* __restrict__ x, const float* __restrict__ ubuf,
               const float* __restrict__ vb, const float* __restrict__ bsrc,
               float* __restrict__ bdst) {
  __shared__ float lu[JJ * LXS];
  const int b = blockIdx.x;
  const int tid = threadIdx.x;

  const float* ub = ubuf + (size_t)b * JJ * CIN;
#pragma unroll
  for (int r = 0; r < 8; ++r) {
    const int flat = r * 1024 + tid * 4;
    const int row = flat >> 8, col = flat & 255;
    *(float4*)&lu[row * LXS + col] = *(const float4*)(ub + row * CIN + col);
  }
  __syncthreads();

  const int w = tid >> 5, L = tid & 31, ln = L & 15;
  const int jt = w & 1, ig = (w >> 1) * 64;  // 4 i-tiles of 16 per wave
  const int koff = (L >> 4) << 1;
  const float* arow = &lu[(jt * 16 + ln) * LXS + koff];
  const float* xb = x + (size_t)b * CIN * II + ln;
  v8f acc[4] = {};
  for (int k0 = 0; k0 < CIN; k0 += 4) {
    const int kb = k0 + koff;
    const v2f a = *(const v2f*)(arow + k0);
    const float* xk = xb + (size_t)kb * II + ig;
    if (k0 + 4 < CIN) __builtin_prefetch(xk + 4 * II, 0, 0);  // next K rows
#pragma unroll
    for (int t = 0; t < 4; ++t) {
      v2f bb;
      bb.x = xk[t * 16];       // B[k,   n=ln] : lanes hit consecutive i
      bb.y = xk[t * 16 + II];  // B[k+1, n=ln]
      acc[t] = __builtin_amdgcn_wmma_f32_16x16x4_f32(false, a, false, bb,
                                                     (short)0, acc[t], false, false);
    }
  }
  const int rowj = jt * 16 + ((L >> 4) << 3);
  const float* bs = bsrc + (size_t)b * JJ * II;
  float* bd = bdst + (size_t)b * JJ * II;
#pragma unroll
  for (int t = 0; t < 4; ++t) {
    const int ib = ig + t * 16 + ln;
#pragma unroll
    for (int r = 0; r < 8; ++r) {
      const int j = rowj + r;
      const size_t idx = (size_t)j * II + ib;
      bd[idx] = bs[idx] + acc[t][r] + vb[b * JJ + j];
    }
  }
}

// ---------------------------------------------------------------------------
extern "C" void kernel_launch(void* const* d_in, const int* in_sizes, int n_in,
                              void* d_out, int out_size, void* d_ws, size_t ws_size,
                              hipStream_t stream) {
  (void)in_sizes; (void)n_in; (void)out_size; (void)ws_size;
  const float* x     = (const float*)d_in[0];  // [BS, C, S, S]
  const float* Wm    = (const float*)d_in[1];  // [J*D, C]
  const float* bias  = (const float*)d_in[2];  // [J*D]
  const float* binit = (const float*)d_in[3];  // [BS, J, I]
  float* vout = (float*)d_out;                 // [BS, J, D]

  float* ws = (float*)d_ws;
  const size_t nBJI = (size_t)BS * JJ * II;
  const size_t nBJC = (size_t)BS * JJ * CIN;
  float* cbuf = ws;                // softmax coefficients [BS,J,I]
  float* yt   = cbuf + nBJI;       // y  [BS,J,C]
  float* ubuf = yt + nBJC;         // u  [BS,J,C]
  float* bbuf = ubuf + nBJC;       // routing logits (updated)
  float* csum = bbuf + nBJI;       // [BS,J]
  float* vb   = csum + (size_t)BS * JJ;  // [BS,J]

  for (int it = 0; it < 3; ++it) {
    const float* bcur = (it == 0) ? binit : bbuf;
    cap_softmax<<<dim3(BS), dim3(256), 0, stream>>>(bcur, cbuf, csum);
    cap_gemm_y<<<dim3(BS, CIN / 64), dim3(256), 0, stream>>>(x, cbuf, yt);
    cap_caps<<<dim3(JJ, BS), dim3(256), 0, stream>>>(yt, Wm, bias, csum,
                                                     vout, ubuf, vb);
    if (it < 2)  // final v comes from iteration 3's squash; its delta is unused
      cap_delta<<<dim3(BS), dim3(256), 0, stream>>>(x, ubuf, vb, bcur, bbuf);
  }
}